// HCCFModel_70085276336651
// MI455X (gfx1250) — compile-verified
//
#include <hip/hip_runtime.h>

#define N_USERS 200000
#define N_ITEMS 400000
#define N_NODES 600000
#define DIMD    32
#define N_HYP   128
#define N_EDGES 10000000

typedef float v2f __attribute__((ext_vector_type(2)));
typedef float v8f __attribute__((ext_vector_type(8)));

__device__ __forceinline__ v8f wmma_f32(v2f a, v2f b, v8f c) {
    // D = A(16x4, f32) * B(4x16, f32) + C(16x16, f32)
    return __builtin_amdgcn_wmma_f32_16x16x4_f32(
        false, a, false, b, (short)0, c, false, false);
}

// ---------------------------------------------------------------------------
// M = W @ W^T  (W is [32,128] row-major), one block per matrix (user/item)
__global__ void k_precompute_M(const float* __restrict__ Wu,
                               const float* __restrict__ Wi,
                               float* __restrict__ Mbase) {
    const float* W = (blockIdx.x == 0) ? Wu : Wi;
    float* M = Mbase + blockIdx.x * (DIMD * DIMD);
    int k = threadIdx.x >> 5;   // 0..31 (blockDim = 1024)
    int j = threadIdx.x & 31;
    float s = 0.f;
#pragma unroll 4
    for (int h = 0; h < N_HYP; ++h)
        s += W[k * N_HYP + h] * W[j * N_HYP + h];
    M[k * DIMD + j] = s;
}

// ---------------------------------------------------------------------------
// buf0 = emb = concat(user_emb, item_emb); out = emb (sequence[0] term)
// buf0 stays regular (it is the next layer's random-gather table -> keep in L2);
// out is a once-per-layer stream -> non-temporal so it doesn't evict tables.
__global__ void k_init(const float* __restrict__ ue, const float* __restrict__ ie,
                       float* __restrict__ buf0, float* __restrict__ out) {
    int idx = blockIdx.x * blockDim.x + threadIdx.x;    // < 19.2M
    const int nu = N_USERS * DIMD;
    float v = (idx < nu) ? __builtin_nontemporal_load(&ue[idx])
                         : __builtin_nontemporal_load(&ie[idx - nu]);
    buf0[idx] = v;
    __builtin_nontemporal_store(v, &out[idx]);
}

// ---------------------------------------------------------------------------
// COO SpMM scatter: temp[row] += val * prev[col]. One wave per edge, lane = dim.
// prev/temp rows are 128B -> one coalesced gather + one cacheline-combined
// atomic per edge; both tables (2 x 76.8MB) live in the 192MB L2.
// Edge arrays (120MB/layer) are touch-once streams -> non-temporal loads.
__global__ void k_gcn_scatter(const int* __restrict__ rows,
                              const int* __restrict__ cols,
                              const float* __restrict__ vals,
                              const float* __restrict__ prev,
                              float* __restrict__ temp) {
    int gid  = blockIdx.x * blockDim.x + threadIdx.x;
    int e    = gid >> 5;
    int lane = gid & 31;
    if (e >= N_EDGES) return;
    int   r = __builtin_nontemporal_load(&rows[e]);
    int   c = __builtin_nontemporal_load(&cols[e]);
    float v = __builtin_nontemporal_load(&vals[e]);
    float x = v * prev[c * DIMD + lane];
    unsafeAtomicAdd(&temp[r * DIMD + lane], x);
}

// ---------------------------------------------------------------------------
// lamb[32,32] += E0^T @ X  (tall reduction over `count` nodes), WMMA f32 16x16x4.
// A[k][n] = E0[n][k], B[n][d] = X[n][d]; K streamed 4 nodes per wmma.
// E0 is a sequential stream -> non-temporal; X (=prev) left regular (L2-hot).
__global__ void k_lamb_wmma(const float* __restrict__ E0,
                            const float* __restrict__ X,
                            float* __restrict__ lamb, int count) {
    int lane = threadIdx.x & 31;
    int half = lane >> 4;       // 0: lanes 0-15, 1: lanes 16-31
    int m    = lane & 15;
    int wavesPerBlock = blockDim.x >> 5;
    int wid    = blockIdx.x * wavesPerBlock + (threadIdx.x >> 5);
    int nWaves = gridDim.x * wavesPerBlock;
    int nChunks = count >> 2;   // 4 nodes per chunk

    v8f c00 = {}, c01 = {}, c10 = {}, c11 = {};
    for (int ch = wid; ch < nChunks; ch += nWaves) {
        int r0 = (ch << 2) + (half << 1);           // node row: nb + 2*half (+v)
        const float* e0 = E0 + r0 * DIMD;
        const float* x0 = X  + r0 * DIMD;
        v2f a0, a1, b0, b1;
        // A layout: lane(m,half), vgpr v -> A[m][2*half+v] = E0[nb+2*half+v][mt*16+m]
        a0.x = __builtin_nontemporal_load(&e0[m]);
        a0.y = __builtin_nontemporal_load(&e0[DIMD + m]);
        a1.x = __builtin_nontemporal_load(&e0[16 + m]);
        a1.y = __builtin_nontemporal_load(&e0[DIMD + 16 + m]);
        // B layout: lane(m,half), vgpr v -> B[2*half+v][nt*16+m] = X[nb+2*half+v][...]
        b0.x = x0[m];        b0.y = x0[DIMD + m];
        b1.x = x0[16 + m];   b1.y = x0[DIMD + 16 + m];
        c00 = wmma_f32(a0, b0, c00);
        c01 = wmma_f32(a0, b1, c01);
        c10 = wmma_f32(a1, b0, c10);
        c11 = wmma_f32(a1, b1, c11);
    }
    // C layout: vgpr r, lane(m,half): row = r + 8*half, col = m
#pragma unroll
    for (int r = 0; r < 8; ++r) {
        int krow = r + (half << 3);
        unsafeAtomicAdd(&lamb[(krow)      * DIMD + m],      c00[r]);
        unsafeAtomicAdd(&lamb[(krow)      * DIMD + 16 + m], c01[r]);
        unsafeAtomicAdd(&lamb[(krow + 16) * DIMD + m],      c10[r]);
        unsafeAtomicAdd(&lamb[(krow + 16) * DIMD + 16 + m], c11[r]);
    }
}

// ---------------------------------------------------------------------------
// inner = M @ lamb  (both 32x32), blockIdx selects user/item
__global__ void k_inner(const float* __restrict__ Mbase,
                        const float* __restrict__ lambbase,
                        float* __restrict__ innerbase) {
    const float* M = Mbase    + blockIdx.x * (DIMD * DIMD);
    const float* L = lambbase + blockIdx.x * (DIMD * DIMD);
    float*       I = innerbase + blockIdx.x * (DIMD * DIMD);
    int k = threadIdx.x >> 5;
    int d = threadIdx.x & 31;
    float s = 0.f;
#pragma unroll
    for (int j = 0; j < DIMD; ++j)
        s += M[k * DIMD + j] * L[j * DIMD + d];
    I[k * DIMD + d] = s;
}

// ---------------------------------------------------------------------------
// Fused: cur = temp + leaky_relu(E0 @ inner, 0.5); out += cur (in-place in temp).
// One wave handles a 16-node x 32-dim tile with 16 WMMA ops.
// E0 and out are touch-once streams -> non-temporal; tempcur stays regular
// (it is the next layer's random-gather table).
__global__ void k_fused_hyp(const float* __restrict__ E0,
                            const float* __restrict__ inner,
                            float* __restrict__ tempcur,
                            float* __restrict__ outp, int count) {
    int lane = threadIdx.x & 31;
    int half = lane >> 4;
    int m    = lane & 15;
    int wid  = blockIdx.x * (blockDim.x >> 5) + (threadIdx.x >> 5);
    int nTiles = count >> 4;
    if (wid >= nTiles) return;          // wave-uniform exit (EXEC stays all-ones)
    int nb = wid << 4;
    const float* arow = E0 + (nb + m) * DIMD;   // node nb+m row (both halves need it)

    v8f c0 = {}, c1 = {};
#pragma unroll
    for (int k0 = 0; k0 < DIMD; k0 += 4) {
        int kk = k0 + (half << 1);
        v2f a, b0, b1;
        a.x  = __builtin_nontemporal_load(&arow[kk]);
        a.y  = __builtin_nontemporal_load(&arow[kk + 1]);
        b0.x = inner[kk * DIMD + m];        b0.y = inner[(kk + 1) * DIMD + m];
        b1.x = inner[kk * DIMD + 16 + m];   b1.y = inner[(kk + 1) * DIMD + 16 + m];
        c0 = wmma_f32(a, b0, c0);
        c1 = wmma_f32(a, b1, c1);
    }
#pragma unroll
    for (int r = 0; r < 8; ++r) {
        int node = nb + r + (half << 3);
        float v0 = c0[r]; v0 = (v0 > 0.f) ? v0 : 0.5f * v0;   // leaky_relu(0.5)
        float v1 = c1[r]; v1 = (v1 > 0.f) ? v1 : 0.5f * v1;
        int i0 = node * DIMD + m;
        int i1 = i0 + 16;
        float s0 = tempcur[i0] + v0;
        float s1 = tempcur[i1] + v1;
        tempcur[i0] = s0;  tempcur[i1] = s1;    // cur for next layer
        float o0 = __builtin_nontemporal_load(&outp[i0]) + s0;
        float o1 = __builtin_nontemporal_load(&outp[i1]) + s1;
        __builtin_nontemporal_store(o0, &outp[i0]);   // running sum of sequence
        __builtin_nontemporal_store(o1, &outp[i1]);
    }
}

// ---------------------------------------------------------------------------
extern "C" void kernel_launch(void* const* d_in, const int* in_sizes, int n_in,
                              void* d_out, int out_size, void* d_ws, size_t ws_size,
                              hipStream_t stream) {
    (void)in_sizes; (void)n_in; (void)out_size; (void)ws_size;
    const int*   adj  = (const int*)d_in[0];
    const int*   rows = adj;
    const int*   cols = adj + N_EDGES;
    const float* vals = (const float*)d_in[1];
    const float* ue   = (const float*)d_in[2];
    const float* ie   = (const float*)d_in[3];
    const float* Wu   = (const float*)d_in[4];
    const float* Wi   = (const float*)d_in[5];
    // d_in[6] = keep_rate == 1 -> dropout is identity, ignored.
    float* out = (float*)d_out;

    const size_t NElems = (size_t)N_NODES * DIMD;
    float* buf0      = (float*)d_ws;
    float* buf1      = buf0 + NElems;
    float* Mbase     = buf1 + NElems;                 // 2 x 32x32
    float* lambbase  = Mbase + 2 * DIMD * DIMD;       // 2 x 32x32
    float* innerbase = lambbase + 2 * DIMD * DIMD;    // 2 x 32x32
    const size_t NB = NElems * sizeof(float);

    k_precompute_M<<<2, 1024, 0, stream>>>(Wu, Wi, Mbase);
    k_init<<<(N_NODES * DIMD) / 256, 256, 0, stream>>>(ue, ie, buf0, out);

    float* prev = buf0;
    float* temp = buf1;
    for (int layer = 0; layer < 2; ++layer) {
        hipMemsetAsync(temp, 0, NB, stream);
        hipMemsetAsync(lambbase, 0, 2 * DIMD * DIMD * sizeof(float), stream);

        k_gcn_scatter<<<(N_EDGES * 32) / 256, 256, 0, stream>>>(rows, cols, vals, prev, temp);

        k_lamb_wmma<<<2048, 256, 0, stream>>>(ue, prev, lambbase, N_USERS);
        k_lamb_wmma<<<2048, 256, 0, stream>>>(ie, prev + (size_t)N_USERS * DIMD,
                                              lambbase + DIMD * DIMD, N_ITEMS);
        k_inner<<<2, 1024, 0, stream>>>(Mbase, lambbase, innerbase);

        k_fused_hyp<<<(N_USERS / 16 + 7) / 8, 256, 0, stream>>>(
            ue, innerbase, temp, out, N_USERS);
        k_fused_hyp<<<(N_ITEMS / 16 + 7) / 8, 256, 0, stream>>>(
            ie, innerbase + DIMD * DIMD, temp + (size_t)N_USERS * DIMD,
            out + (size_t)N_USERS * DIMD, N_ITEMS);

        float* t = prev; prev = temp; temp = t;   // ping-pong node buffers
    }
}